// Experts_45956150067877
// MI455X (gfx1250) — compile-verified
//
#include <hip/hip_runtime.h>
#include <hip/hip_bf16.h>
#include <stdint.h>

typedef __attribute__((ext_vector_type(2))) float v2f;
typedef __attribute__((ext_vector_type(4))) float v4f;
typedef __attribute__((ext_vector_type(8))) float v8f;

#define NUM_EXPERTS 8
#define HIDDEN     2048
#define INTER      6144
#define TPE        1024   // tokens per expert

// Tiling: workgroup = 256 threads = 8 waves, computes TM x TN output tile.
// Each wave owns a 16 x TN strip. K is staged through LDS in TK chunks.
#define TM 128
#define TN 64
#define TK 32
// A tile: [TM][TK] floats at row stride LXS. Even stride -> 8B-aligned K-pairs
// (ds_load_b64); banks 36r mod 64 distinct for 16 rows; cross-half conflict
// would need 18*dr == -1 (mod 32): impossible.
#define LXS 36
// B tile: K-pair interleaved: elem(k,n) at (k>>1)*BKS + 2n + (k&1).
// BKS = 2*TN + 32 = 160 -> +1 k-pair lands +32 banks: half-waves disjoint.
#define BKS 160

__device__ __forceinline__ v8f wmma4(v2f a, v2f b, v8f c) {
    // D = A(16x4,f32) * B(4x16,f32) + C(16x16,f32)
    return __builtin_amdgcn_wmma_f32_16x16x4_f32(
        /*neg_a=*/false, a, /*neg_b=*/false, b,
        /*c_mod=*/(short)0, c, /*reuse_a=*/false, /*reuse_b=*/false);
}

// Async global -> LDS copy, 16B per lane (GLOBAL_LOAD_ASYNC_TO_LDS_B128, GV
// mode). lds_off = per-lane LDS byte offset (generic shared addr truncated to
// 32 bits: LDS aperture keeps the offset in addr[31:0]). Tracked on ASYNCcnt.
__device__ __forceinline__ void async_copy_b128(uint32_t lds_off, const void* g) {
    asm volatile("global_load_async_to_lds_b128 %0, %1, off"
                 :: "v"(lds_off), "v"(g)
                 : "memory");
}
__device__ __forceinline__ void wait_async0() {
    asm volatile("s_wait_asynccnt 0x0" ::: "memory");
}

// ---------------------------------------------------------------------------
// Kernel 1: hidden = silu(x @ Wgate) * (x @ Wup)   per expert
//   x   : [E*TPE, HIDDEN]          row-major
//   wgu : [E, HIDDEN, 2*INTER]     row-major (gate cols [0,INTER), up [INTER,2I))
//   h   : [E*TPE, INTER]           row-major (workspace)
// grid: (INTER/TN, TPE/TM, E)
// ---------------------------------------------------------------------------
__global__ __launch_bounds__(256)
void moe_gateup_silu(const float* __restrict__ x,
                     const float* __restrict__ wgu,
                     float* __restrict__ h)
{
    __shared__ float lx [TM * LXS];            // 18.0 KB
    __shared__ float lwg[(TK / 2) * BKS];      // 10.0 KB
    __shared__ float lwu[(TK / 2) * BKS];      // 10.0 KB

    const int n0 = blockIdx.x * TN;
    const int t0 = blockIdx.y * TM;
    const int e  = blockIdx.z;

    const int tid  = threadIdx.x;
    const int lane = tid & 31;
    const int wv   = tid >> 5;       // wave id 0..7 -> M sub-strip
    const int half = lane >> 4;      // half-wave (K pair select in fragments)
    const int ml   = lane & 15;      // M (A frag) / N (B,C frags) within 16

    const float* xe = x   + (size_t)(e * TPE + t0) * HIDDEN;
    const float* wg = wgu + (size_t)e * HIDDEN * (2 * INTER) + n0;
    const float* wu = wg + INTER;

    const uint32_t lxoff = (uint32_t)(uintptr_t)&lx[0];

    const v8f vzero = {0.f,0.f,0.f,0.f,0.f,0.f,0.f,0.f};
    v8f accg[4], accu[4];
#pragma unroll
    for (int j = 0; j < 4; ++j) { accg[j] = vzero; accu[j] = vzero; }

    const int arow = (wv * 16 + ml) * LXS;

    for (int k0 = 0; k0 < HIDDEN; k0 += TK) {
        // ---- stage A tile async: 128 rows x 32 k (1024 x 16B, 4 per lane) ----
#pragma unroll
        for (int it = 0; it < 4; ++it) {
            const int idx = tid + it * 256;
            const int row = idx >> 3;
            const int c4  = (idx & 7) << 2;
            async_copy_b128(lxoff + (uint32_t)(row * LXS + c4) * 4u,
                            xe + (size_t)row * HIDDEN + k0 + c4);
        }
        // ---- stage B tiles (gate & up): 32 k x 64 n, k-pair interleaved ----
#pragma unroll
        for (int it = 0; it < 2; ++it) {
            const int idx = tid + it * 256;
            const int row = idx >> 4;           // k within chunk
            const int c4  = (idx & 15) << 2;    // n base
            const size_t off = (size_t)(k0 + row) * (2 * INTER) + c4;
            const v4f g = __builtin_nontemporal_load((const v4f*)(wg + off));
            const v4f u = __builtin_nontemporal_load((const v4f*)(wu + off));
            float* dg = &lwg[(row >> 1) * BKS + 2 * c4 + (row & 1)];
            float* du = &lwu[(row >> 1) * BKS + 2 * c4 + (row & 1)];
            dg[0] = g.x; dg[2] = g.y; dg[4] = g.z; dg[6] = g.w;
            du[0] = u.x; du[2] = u.y; du[4] = u.z; du[6] = u.w;
            if (k0 + TK < HIDDEN) {
                __builtin_prefetch(wg + off + (size_t)TK * (2 * INTER), 0, 0);
                __builtin_prefetch(wu + off + (size_t)TK * (2 * INTER), 0, 0);
            }
        }
        wait_async0();          // A-tile async copies landed in LDS
        __syncthreads();

        // ---- 8 WMMA K-steps over the staged chunk ----
#pragma unroll
        for (int kk = 0; kk < TK; kk += 4) {
            const int kp = (kk >> 1) + half;    // this half-wave's k-pair
            const v2f a = *(const v2f*)(&lx[arow + kk + half * 2]);
#pragma unroll
            for (int j = 0; j < 4; ++j) {
                const v2f bg = *(const v2f*)(&lwg[kp * BKS + 2 * (j * 16 + ml)]);
                const v2f bu = *(const v2f*)(&lwu[kp * BKS + 2 * (j * 16 + ml)]);
                accg[j] = wmma4(a, bg, accg[j]);
                accu[j] = wmma4(a, bu, accu[j]);
            }
        }
        __syncthreads();
    }

    // ---- epilogue: silu(gate) * up, scatter per C/D fragment layout ----
    float* hout = h + (size_t)(e * TPE + t0 + wv * 16 + half * 8) * INTER + n0 + ml;
#pragma unroll
    for (int j = 0; j < 4; ++j) {
#pragma unroll
        for (int r = 0; r < 8; ++r) {
            const float g = accg[j][r];
            const float u = accu[j][r];
            const float s = g / (1.0f + __expf(-g));
            hout[(size_t)r * INTER + j * 16] = s * u;
        }
    }
}

// ---------------------------------------------------------------------------
// Kernel 2: out = h @ Wdown   per expert
//   h   : [E*TPE, INTER]
//   wd  : [E, INTER, HIDDEN]
//   out : [E*TPE, HIDDEN]
// grid: (HIDDEN/TN, TPE/TM, E)
// ---------------------------------------------------------------------------
__global__ __launch_bounds__(256)
void moe_down(const float* __restrict__ h,
              const float* __restrict__ wd,
              float* __restrict__ out)
{
    __shared__ float lh[TM * LXS];
    __shared__ float lw[(TK / 2) * BKS];

    const int n0 = blockIdx.x * TN;
    const int t0 = blockIdx.y * TM;
    const int e  = blockIdx.z;

    const int tid  = threadIdx.x;
    const int lane = tid & 31;
    const int wv   = tid >> 5;
    const int half = lane >> 4;
    const int ml   = lane & 15;

    const float* he = h  + (size_t)(e * TPE + t0) * INTER;
    const float* we = wd + (size_t)e * INTER * HIDDEN + n0;

    const uint32_t lhoff = (uint32_t)(uintptr_t)&lh[0];

    const v8f vzero = {0.f,0.f,0.f,0.f,0.f,0.f,0.f,0.f};
    v8f acc[4];
#pragma unroll
    for (int j = 0; j < 4; ++j) acc[j] = vzero;

    const int arow = (wv * 16 + ml) * LXS;

    for (int k0 = 0; k0 < INTER; k0 += TK) {
        // ---- stage A tile async: 128 x 32 from hidden ----
#pragma unroll
        for (int it = 0; it < 4; ++it) {
            const int idx = tid + it * 256;
            const int row = idx >> 3;
            const int c4  = (idx & 7) << 2;
            async_copy_b128(lhoff + (uint32_t)(row * LXS + c4) * 4u,
                            he + (size_t)row * INTER + k0 + c4);
        }
        // ---- stage B tile: 32 k x 64 n from w_down, k-pair interleaved ----
#pragma unroll
        for (int it = 0; it < 2; ++it) {
            const int idx = tid + it * 256;
            const int row = idx >> 4;
            const int c4  = (idx & 15) << 2;
            const size_t off = (size_t)(k0 + row) * HIDDEN + c4;
            const v4f w = __builtin_nontemporal_load((const v4f*)(we + off));
            float* dw = &lw[(row >> 1) * BKS + 2 * c4 + (row & 1)];
            dw[0] = w.x; dw[2] = w.y; dw[4] = w.z; dw[6] = w.w;
            if (k0 + TK < INTER)
                __builtin_prefetch(we + off + (size_t)TK * HIDDEN, 0, 0);
        }
        wait_async0();
        __syncthreads();

#pragma unroll
        for (int kk = 0; kk < TK; kk += 4) {
            const int kp = (kk >> 1) + half;
            const v2f a = *(const v2f*)(&lh[arow + kk + half * 2]);
#pragma unroll
            for (int j = 0; j < 4; ++j) {
                const v2f b = *(const v2f*)(&lw[kp * BKS + 2 * (j * 16 + ml)]);
                acc[j] = wmma4(a, b, acc[j]);
            }
        }
        __syncthreads();
    }

    float* oo = out + (size_t)(e * TPE + t0 + wv * 16 + half * 8) * HIDDEN + n0 + ml;
#pragma unroll
    for (int j = 0; j < 4; ++j) {
#pragma unroll
        for (int r = 0; r < 8; ++r) {
            oo[(size_t)r * HIDDEN + j * 16] = acc[j][r];
        }
    }
}

// ---------------------------------------------------------------------------
extern "C" void kernel_launch(void* const* d_in, const int* in_sizes, int n_in,
                              void* d_out, int out_size, void* d_ws, size_t ws_size,
                              hipStream_t stream)
{
    const float* x   = (const float*)d_in[0];   // routed_tokens [E*TPE, HIDDEN]
    const float* wgu = (const float*)d_in[1];   // [E, HIDDEN, 2*INTER]
    const float* wd  = (const float*)d_in[2];   // [E, INTER, HIDDEN]
    float* out = (float*)d_out;                 // [E*TPE, HIDDEN]
    float* hws = (float*)d_ws;                  // needs E*TPE*INTER*4 = 192 MiB

    dim3 blk(256);
    moe_gateup_silu<<<dim3(INTER / TN, TPE / TM, NUM_EXPERTS), blk, 0, stream>>>(x, wgu, hws);
    moe_down      <<<dim3(HIDDEN / TN, TPE / TM, NUM_EXPERTS), blk, 0, stream>>>(hws, wd, out);
}